// Entmax15Explainer_27161373180348
// MI455X (gfx1250) — compile-verified
//
#include <hip/hip_runtime.h>

// CDNA5 / gfx1250: wave32, WMMA f32 16x16x4
typedef __attribute__((ext_vector_type(2))) float v2f;
typedef __attribute__((ext_vector_type(8))) float v8f;

#define B_      64
#define L_      12
#define H_      12
#define S_      8192
#define THREADS 1024
#define WAVES   (THREADS / 32)
#define PER_THR (S_ / THREADS)   // 8 elements per thread held in registers

__global__ __launch_bounds__(THREADS)
void entmax15_kernel(const float* __restrict__ attn,
                     float* __restrict__ out_p,
                     float* __restrict__ out_logits) {
    __shared__ float zbuf[S_];     // 32 KB: row logits staging
    __shared__ float wred[WAVES];  // per-wave reduction slots

    const int tid  = threadIdx.x;
    const int wid  = tid >> 5;
    const int lane = tid & 31;
    const int half = lane >> 4;    // which half-wave (selects K rows of B)
    const int n    = lane & 15;    // N column inside the 16-wide tile
    const int b    = blockIdx.x;

    // attention layout [B, L, H, 1, S]; slice layer L-1
    const float* __restrict__ row =
        attn + ((size_t)b * L_ + (L_ - 1)) * (size_t)H_ * S_;
    float* __restrict__ logits_out = out_logits + (size_t)b * S_;

    // ---- Phase 1: mean over 12 heads via 3 chained V_WMMA_F32_16X16X4_F32 ----
    // A = constant 16x4 matrix of 1/12 (splat -> layout independent).
    v2f aOnes; aOnes.x = 1.0f / 12.0f; aOnes.y = 1.0f / 12.0f;

    for (int tile = wid; tile < S_ / 16; tile += WAVES) {
        const int s = tile * 16 + n;
        v8f c = {};
#pragma unroll
        for (int hb = 0; hb < H_; hb += 4) {
            const int h0 = hb + half * 2;           // 2 K-rows per half-wave
            v2f bm;
            bm.x = row[(size_t)h0 * S_ + s];        // K = half*2
            bm.y = row[(size_t)(h0 + 1) * S_ + s];  // K = half*2 + 1
            // D = A(16x4, all 1/12) x B(4x16 heads-by-s) + C
            c = __builtin_amdgcn_wmma_f32_16x16x4_f32(
                    false, aOnes, false, bm, (short)0, c, false, false);
        }
        // every row of D is identical; c[0] = mean at column (lane&15)
        const float mean = c[0];
        if (half == 0) {
            zbuf[s]       = mean;
            logits_out[s] = mean;   // second tuple output: attention_logits
        }
    }
    __syncthreads();

    // ---- Phase 2: load my 8 elements into registers, block max ----
    float r[PER_THR];
    float lmax = -3.402823466e+38f;
#pragma unroll
    for (int k = 0; k < PER_THR; ++k) {
        r[k] = zbuf[tid + k * THREADS];
        lmax = fmaxf(lmax, r[k]);
    }
#pragma unroll
    for (int i = 16; i >= 1; i >>= 1)
        lmax = fmaxf(lmax, __shfl_xor(lmax, i, 32));
    if (lane == 0) wred[wid] = lmax;
    __syncthreads();
    float gmax = wred[0];
#pragma unroll
    for (int w = 1; w < WAVES; ++w) gmax = fmaxf(gmax, wred[w]);

    // z = logits/2 - max(logits/2)  ==  (logits - max)/2 ; now max(z) == 0
#pragma unroll
    for (int k = 0; k < PER_THR; ++k)
        r[k] = (r[k] - gmax) * 0.5f;

    // ---- Phase 3: tau via bisection of f(tau)=sum clip(z-tau,0)^2 - 1 ----
    // f is strictly decreasing; root bracketed in [-1, 0] after the shift.
    float lo = -1.0f, hi = 0.0f;
    for (int it = 0; it < 40; ++it) {
        const float mid = 0.5f * (lo + hi);
        float s = 0.0f;
#pragma unroll
        for (int k = 0; k < PER_THR; ++k) {
            float t = fmaxf(r[k] - mid, 0.0f);
            s = fmaf(t, t, s);
        }
#pragma unroll
        for (int i = 16; i >= 1; i >>= 1)
            s += __shfl_xor(s, i, 32);
        __syncthreads();                 // protect wred reuse
        if (lane == 0) wred[wid] = s;
        __syncthreads();
        float S = 0.0f;
#pragma unroll
        for (int w = 0; w < WAVES; ++w) S += wred[w];   // identical in all threads
        if (S >= 1.0f) lo = mid; else hi = mid;          // uniform branch
    }
    const float tau = 0.5f * (lo + hi);

    // ---- Phase 4: p = clip(z - tau, 0)^2, coalesced store ----
    float* __restrict__ p_out = out_p + (size_t)b * S_;
#pragma unroll
    for (int k = 0; k < PER_THR; ++k) {
        float t = fmaxf(r[k] - tau, 0.0f);
        p_out[tid + k * THREADS] = t * t;
    }
}

extern "C" void kernel_launch(void* const* d_in, const int* in_sizes, int n_in,
                              void* d_out, int out_size, void* d_ws, size_t ws_size,
                              hipStream_t stream) {
    const float* attn = (const float*)d_in[0];
    float* out        = (float*)d_out;
    float* out_p      = out;                         // first tuple element
    float* out_logits = out + (size_t)B_ * S_;       // second tuple element
    entmax15_kernel<<<B_, THREADS, 0, stream>>>(attn, out_p, out_logits);
}